// FSRS2_31585189494794
// MI455X (gfx1250) — compile-verified
//
#include <hip/hip_runtime.h>

#define FSRS_S_MIN 0.01f
#define FSRS_S_MAX 36500.0f

__global__ __launch_bounds__(256) void fsrs2_scan_kernel(
    const float* __restrict__ inputs,    // [T, B, 2]
    const float* __restrict__ delta_ts,  // [B]
    const float* __restrict__ w,         // [14]
    const int*   __restrict__ seq_lens,  // [B]
    float*       __restrict__ out,       // [3, B]
    int batch)
{
    const int b = blockIdx.x * blockDim.x + threadIdx.x;
    if (b >= batch) return;

    // Uniform weight loads at constant offsets -> scalarized to s_load_b256.
    const float w0 = w[0],  w1 = w[1],  w2 = w[2],  w3 = w[3],  w4 = w[4];
    const float w5 = w[5],  w6 = w[6],  w7 = w[7],  w8 = w[8],  w9 = w[9];
    const float w10 = w[10], w11 = w[11], w12 = w[12], w13 = w[13];

    const float LOG2E = 1.4426950408889634f;
    const float L09   = -0.15200309344504997f;  // log2(0.9)
    const float C6   = w6  * LOG2E;             // exp(w6)   -> exp2(C6)
    const float C9   = w9  * LOG2E;             // exp(x*w9) -> exp2(x*C9)
    const float C13  = w13 * LOG2E;
    const float a_nd = w5 * (w2 * (1.0f - w3)); // mean-reversion target term
    const float b_nd = 1.0f - w5;

    const float2* __restrict__ in2 = reinterpret_cast<const float2*>(inputs);

    const int L = seq_lens[b];

    // ---- t = 0 (peeled "first" branch) ----
    const float2 X0 = in2[b];
    const float rating0 = X0.y;
    float s = fminf(fmaxf(w0 * fmaf(w1, rating0 - 1.0f, 1.0f), FSRS_S_MIN), FSRS_S_MAX);
    float d = fminf(fmaxf(w2 * fmaf(w3, rating0 - 4.0f, 1.0f), 1.0f), 10.0f);

    // ---- t = 1 .. L-1, software-pipelined loads, branchless prefetch ----
    if (L > 1) {
        const int Lm1 = L - 1;
        float2 Xc = in2[(size_t)batch + b];  // t = 1
        for (int t = 1; t < L; ++t) {
            const int tn = (t + 1 < L) ? (t + 1) : t;          // clamped: always valid
            const float2 Xn = in2[(size_t)tn * batch + b];     // next-iter data in flight
            const int tp = (t + 6 < Lm1) ? (t + 6) : Lm1;      // clamped, unconditional
            __builtin_prefetch(&in2[(size_t)tp * batch + b], 0, 0); // global_prefetch_b8

            const float elapsed = Xc.x;
            const float rating  = Xc.y;

            const float ss  = fmaxf(s, FSRS_S_MIN);
            const float lss = __builtin_amdgcn_logf(ss);                         // v_log_f32
            const float r   = __builtin_amdgcn_exp2f(L09 * elapsed *
                                                     __builtin_amdgcn_rcpf(ss)); // 0.9^(elapsed/ss)

            float nd = fmaf(w4, rating - 3.0f, d);
            nd = fmaf(b_nd, nd, a_nd);
            nd = fminf(fmaxf(nd, 1.0f), 10.0f);
            const float lnd = __builtin_amdgcn_logf(nd);

            const float omr  = 1.0f - r;
            const float gain = __builtin_amdgcn_exp2f(omr * C9) - 1.0f;
            // exp(w6) * nd^w7 * ss^w8 fused into one exp2
            const float es   = __builtin_amdgcn_exp2f(fmaf(w7, lnd, fmaf(w8, lss, C6)));
            const float succ = ss * fmaf(es, gain, 1.0f);

            // w10 * nd^w11 * ss^w12 * (exp((1-r)*w13) - 1), pow pair fused
            const float ef   = __builtin_amdgcn_exp2f(fmaf(w11, lnd, w12 * lss));
            const float gf   = __builtin_amdgcn_exp2f(omr * C13) - 1.0f;
            const float fail = w10 * ef * gf;

            const float ns = (rating > 1.0f) ? succ : fail;
            s = fminf(fmaxf(ns, FSRS_S_MIN), FSRS_S_MAX);
            d = nd;
            Xc = Xn;
        }
    }

    // ---- epilogue: retention = 0.9^(delta_t / s) ----
    const float ret = __builtin_amdgcn_exp2f(L09 * delta_ts[b] * __builtin_amdgcn_rcpf(s));
    out[b]             = ret;
    out[batch + b]     = s;
    out[2 * batch + b] = d;
}

extern "C" void kernel_launch(void* const* d_in, const int* in_sizes, int n_in,
                              void* d_out, int out_size, void* d_ws, size_t ws_size,
                              hipStream_t stream) {
    const float* inputs   = (const float*)d_in[0];  // [64, B, 2] f32
    const float* delta_ts = (const float*)d_in[1];  // [B] f32
    const float* w        = (const float*)d_in[2];  // [14] f32
    const int*   seq_lens = (const int*)d_in[3];    // [B] i32

    float* out = (float*)d_out;                     // [3, B] f32
    const int batch = in_sizes[1];                  // delta_ts element count

    const int threads = 256;                        // 8 wave32 per block
    const int blocks  = (batch + threads - 1) / threads;
    hipLaunchKernelGGL(fsrs2_scan_kernel, dim3(blocks), dim3(threads), 0, stream,
                       inputs, delta_ts, w, seq_lens, out, batch);
}